// GCNLayer_27650999452124
// MI455X (gfx1250) — compile-verified
//
#include <hip/hip_runtime.h>
#include <hip/hip_bf16.h>

// ---------------------------------------------------------------------------
// GCN layer for MI455X (gfx1250, wave32, WMMA, async-LDS)
//   B=16, S=512, D=U=512, L=64, DEG=5
//   Pass 1: Y_{in,out,self} = X @ {V_in, V_out, W_self}  (fp32 WMMA 16x16x4,
//           X tile moved via GLOBAL_LOAD_ASYNC_TO_LDS_B128, W tile stored
//           transposed in LDS so B fragments are single ds_load_b64)
//   Pass 2: gate GEMVs (one wave per row)
//   Pass 3: gather/combine with sigmoid gates, ReLU, transpose to [S,B,U]
// ---------------------------------------------------------------------------

#define Bsz 16
#define Ssz 512
#define Dd  512
#define Uu  512
#define Ll  64
#define DEG 5
#define Nn  (Bsz * Ssz)   // 8192
#define Ee  (Nn * DEG)    // 40960

typedef __attribute__((ext_vector_type(2))) float v2f;
typedef __attribute__((ext_vector_type(8))) float v8f;

// ---- tiled fp32 WMMA GEMM: Y[z] = X @ W[z], X row r = src[(r%S)*B + r/S] ----
#define TM 128
#define TN 128
#define TK 32
#define LSTRIDE 36   // 32-float row + 4 pad: 144B rows (16B aligned, bank-spread)

__global__ __launch_bounds__(256) void gemm_wmma_f32(
    const float* __restrict__ src,
    const float* __restrict__ Vin,
    const float* __restrict__ Vout,
    const float* __restrict__ Wself,
    float* __restrict__ Ybase)
{
    __shared__ float As[TM * LSTRIDE];   // X tile:    [row][k] 128x32 (+pad)
    __shared__ float Bs[TN * LSTRIDE];   // W tile^T:  [n][k]   128x32 (+pad)

    const int z = blockIdx.z;
    const float* __restrict__ W = (z == 0) ? Vin : ((z == 1) ? Vout : Wself);
    float* __restrict__ Y = Ybase + (size_t)z * Nn * Uu;

    const int n0     = blockIdx.x * TN;
    const int m_base = blockIdx.y * TM;
    const int tid  = threadIdx.x;
    const int lane = tid & 31;
    const int wid  = tid >> 5;       // 8 waves
    const int half = lane >> 4;      // 0: lanes 0-15, 1: lanes 16-31
    const int lm   = lane & 15;
    const int m0   = wid * 16;       // wave's 16-row strip inside tile

    const int nIdx = tid & 127;      // W-tile fill: column owned by this thread
    const int kGrp = (tid >> 7) << 4;    // 0 or 16

    v8f acc[8] = {};

    for (int k0 = 0; k0 < Dd; k0 += TK) {
        __syncthreads();

        // ---- X tile via async global->LDS DMA (16B per lane-op, ASYNCcnt) ----
        #pragma unroll
        for (int i = 0; i < 4; ++i) {
            int slot = tid + i * 256;          // 1024 float4 slots
            int row  = slot >> 3;              // 0..127
            int kk4  = (slot & 7) << 2;        // 0..28
            int rg   = m_base + row;           // node id = b*S + s
            int b    = rg >> 9;                // / S
            int s    = rg & (Ssz - 1);
            const float* gp = src + (size_t)(s * Bsz + b) * Dd + k0 + kk4;
            unsigned lds_off =
                (unsigned)(uintptr_t)(const void*)(As + row * LSTRIDE + kk4);
            asm volatile("global_load_async_to_lds_b128 %0, %1, off"
                         :: "v"(lds_off), "v"(gp) : "memory");
        }

        // ---- W tile -> transposed LDS [n][k]; coalesced loads, bank-free stores
        #pragma unroll
        for (int kk = 0; kk < 16; ++kk) {
            Bs[nIdx * LSTRIDE + kGrp + kk] =
                W[(size_t)(k0 + kGrp + kk) * Uu + n0 + nIdx];
        }

        asm volatile("s_wait_asynccnt 0x0" ::: "memory");
        __syncthreads();

        // ---- 8 k-steps of 4, 8 WMMA tiles along N per step ----
        #pragma unroll
        for (int kk = 0; kk < TK; kk += 4) {
            const int ka = kk + 2 * half;      // lanes 0-15: K=kk,kk+1 ; 16-31: kk+2,kk+3
            v2f a = *(const v2f*)(As + (m0 + lm) * LSTRIDE + ka);
            #pragma unroll
            for (int t = 0; t < 8; ++t) {
                v2f b2 = *(const v2f*)(Bs + (t * 16 + lm) * LSTRIDE + ka);
                acc[t] = __builtin_amdgcn_wmma_f32_16x16x4_f32(
                    /*neg_a=*/false, a, /*neg_b=*/false, b2,
                    /*c_mod=*/(short)0, acc[t],
                    /*reuse_a=*/false, /*reuse_b=*/false);
            }
        }
    }

    // ---- store 16x128 strip: VGPR j -> row m0+j (lanes 0-15) / m0+8+j (16-31)
    #pragma unroll
    for (int t = 0; t < 8; ++t) {
        const int n = n0 + t * 16 + lm;
        #pragma unroll
        for (int j = 0; j < 8; ++j) {
            const int m = m_base + m0 + half * 8 + j;
            Y[(size_t)m * Uu + n] = acc[t][j];
        }
    }
}

// ---- gate GEMVs: g[z][r] = dot(X_row_r, Vgate[z]) -------------------------
__global__ __launch_bounds__(32) void gate_dot(
    const float* __restrict__ src,
    const float* __restrict__ g0,
    const float* __restrict__ g1,
    const float* __restrict__ g2,
    float* __restrict__ gates)
{
    const int r = blockIdx.x;
    const int z = blockIdx.y;
    const float* __restrict__ V = (z == 0) ? g0 : ((z == 1) ? g1 : g2);
    const int b = r >> 9, s = r & (Ssz - 1);
    const float* __restrict__ xrow = src + (size_t)(s * Bsz + b) * Dd;
    const int lane = threadIdx.x;
    float sum = 0.f;
    #pragma unroll 4
    for (int d = lane; d < Dd; d += 32) sum += xrow[d] * V[d];
    #pragma unroll
    for (int off = 16; off; off >>= 1) sum += __shfl_xor(sum, off, 32);
    if (lane == 0) gates[(size_t)z * Nn + r] = sum;
}

// ---- gather / gate / reduce / transpose -----------------------------------
__global__ __launch_bounds__(128) void combine(
    const float* __restrict__ Yin, const float* __restrict__ Yout,
    const float* __restrict__ Ysame,
    const float* __restrict__ gin, const float* __restrict__ gout,
    const float* __restrict__ gsame,
    const int* __restrict__ arc_in, const int* __restrict__ arc_out,
    const int* __restrict__ lab_in, const int* __restrict__ lab_out,
    const float* __restrict__ b_in, const float* __restrict__ b_out,
    const float* __restrict__ b_in_gate, const float* __restrict__ b_out_gate,
    const float* __restrict__ mask_in, const float* __restrict__ mask_out,
    const float* __restrict__ mask_loop, const float* __restrict__ sent_mask,
    float* __restrict__ out)
{
    __shared__ float p[11];
    __shared__ int   gidx[10];
    __shared__ int   glab[10];

    const int n   = blockIdx.x;
    const int tid = threadIdx.x;

    if (tid < DEG) {                              // in-arc probs
        const int e   = n * DEG + tid;
        const int idx = arc_in[e] * Ssz + arc_in[Ee + e];
        const int lab = lab_in[e];
        gidx[tid] = idx; glab[tid] = lab;
        const float g = gin[idx] + b_in_gate[lab];
        p[tid] = (1.f / (1.f + __expf(-g))) * mask_in[e];
    } else if (tid < 2 * DEG) {                   // out-arc probs
        const int k   = tid - DEG;
        const int e   = n * DEG + k;
        const int idx = arc_out[e] * Ssz + arc_out[Ee + e];
        const int lab = lab_out[e];
        gidx[tid] = idx; glab[tid] = lab;
        const float g = gout[idx] + b_out_gate[lab];
        p[tid] = (1.f / (1.f + __expf(-g))) * mask_out[e];
    } else if (tid == 2 * DEG) {                  // self-loop prob
        p[10] = (1.f / (1.f + __expf(-gsame[n]))) * mask_loop[n];
    }
    __syncthreads();

    const int u0 = tid * 4;                       // 128 threads x float4 = U
    float ax = 0.f, ay = 0.f, az = 0.f, aw = 0.f;

    #pragma unroll
    for (int k = 0; k < DEG; ++k) {
        const float pk  = p[k];
        const int   row = gidx[k];
        const int   lab = glab[k];
        const float4 y  = *(const float4*)(Yin  + (size_t)row * Uu + u0);
        const float4 bb = *(const float4*)(b_in + (size_t)lab * Uu + u0);
        ax += pk * (y.x + bb.x); ay += pk * (y.y + bb.y);
        az += pk * (y.z + bb.z); aw += pk * (y.w + bb.w);
    }
    #pragma unroll
    for (int k = 0; k < DEG; ++k) {
        const float pk  = p[DEG + k];
        const int   row = gidx[DEG + k];
        const int   lab = glab[DEG + k];
        const float4 y  = *(const float4*)(Yout  + (size_t)row * Uu + u0);
        const float4 bb = *(const float4*)(b_out + (size_t)lab * Uu + u0);
        ax += pk * (y.x + bb.x); ay += pk * (y.y + bb.y);
        az += pk * (y.z + bb.z); aw += pk * (y.w + bb.w);
    }
    {
        const float ps = p[10];
        const float4 y = *(const float4*)(Ysame + (size_t)n * Uu + u0);
        ax += ps * y.x; ay += ps * y.y; az += ps * y.z; aw += ps * y.w;
    }

    ax = fmaxf(ax, 0.f); ay = fmaxf(ay, 0.f);
    az = fmaxf(az, 0.f); aw = fmaxf(aw, 0.f);

    const int b = n >> 9, s = n & (Ssz - 1);
    const float sm = sent_mask[(size_t)s * Bsz + b];
    float4 r; r.x = ax * sm; r.y = ay * sm; r.z = az * sm; r.w = aw * sm;
    *(float4*)(out + ((size_t)s * Bsz + b) * Uu + u0) = r;
}

// ---------------------------------------------------------------------------
extern "C" void kernel_launch(void* const* d_in, const int* in_sizes, int n_in,
                              void* d_out, int out_size, void* d_ws, size_t ws_size,
                              hipStream_t stream) {
    const float* src        = (const float*)d_in[0];
    const int*   arc_in     = (const int*)  d_in[1];
    const int*   arc_out    = (const int*)  d_in[2];
    const int*   lab_in     = (const int*)  d_in[3];
    const int*   lab_out    = (const int*)  d_in[4];
    const float* mask_in    = (const float*)d_in[5];
    const float* mask_out   = (const float*)d_in[6];
    const float* mask_loop  = (const float*)d_in[7];
    const float* sent_mask  = (const float*)d_in[8];
    const float* V_in       = (const float*)d_in[9];
    const float* b_in       = (const float*)d_in[10];
    const float* V_in_gate  = (const float*)d_in[11];
    const float* b_in_gate  = (const float*)d_in[12];
    const float* V_out      = (const float*)d_in[13];
    const float* b_out      = (const float*)d_in[14];
    const float* V_out_gate = (const float*)d_in[15];
    const float* b_out_gate = (const float*)d_in[16];
    const float* W_self     = (const float*)d_in[17];
    const float* W_self_g   = (const float*)d_in[18];

    float* ws    = (float*)d_ws;
    float* Y     = ws;                          // 3 * N * U floats (48 MB)
    float* gates = ws + (size_t)3 * Nn * Uu;    // 3 * N floats
    float* outp  = (float*)d_out;

    // Pass 1: three fused-by-grid.z WMMA GEMMs
    gemm_wmma_f32<<<dim3(Uu / TN, Nn / TM, 3), 256, 0, stream>>>(
        src, V_in, V_out, W_self, Y);

    // Pass 2: gate GEMVs
    gate_dot<<<dim3(Nn, 3), 32, 0, stream>>>(
        src, V_in_gate, V_out_gate, W_self_g, gates);

    // Pass 3: gather + gate + reduce + transpose
    combine<<<Nn, 128, 0, stream>>>(
        Y, Y + (size_t)Nn * Uu, Y + (size_t)2 * Nn * Uu,
        gates, gates + Nn, gates + (size_t)2 * Nn,
        arc_in, arc_out, lab_in, lab_out,
        b_in, b_out, b_in_gate, b_out_gate,
        mask_in, mask_out, mask_loop, sent_mask, outp);
}